// Seq2seq_attn_84310208021000
// MI455X (gfx1250) — compile-verified
//
#include <hip/hip_runtime.h>
#include <hip/hip_bf16.h>
#include <math.h>

// ---------------------------------------------------------------------------
// Problem constants (match reference)
// ---------------------------------------------------------------------------
static constexpr int B_    = 1024;
static constexpr int S_IN  = 96;
static constexpr int S_PR  = 48;
static constexpr int F_    = 512;
static constexpr int H_    = 256;
static constexpr int M_ENC = B_ * S_IN;   // 98304 rows

typedef __attribute__((ext_vector_type(2))) float v2f;
typedef __attribute__((ext_vector_type(8))) float v8f;

// Native FP32 matrix op: D(16x16,f32) = A(16x4,f32) * B(4x16,f32) + C
__device__ __forceinline__ v8f wmma4(v2f a, v2f b, v8f c) {
    return __builtin_amdgcn_wmma_f32_16x16x4_f32(
        /*neg_a=*/false, a, /*neg_b=*/false, b,
        /*c_mod=*/(short)0, c, /*reuse_a=*/false, /*reuse_b=*/false);
}

__device__ __forceinline__ float sigmoidf_(float x) {
    return 1.0f / (1.0f + __expf(-x));
}

template<int ACT>
__device__ __forceinline__ float activate_(float x) {
    if (ACT == 1) return fmaxf(x, 0.0f);   // relu
    if (ACT == 2) return tanhf(x);         // tanh
    return x;
}

// ---------------------------------------------------------------------------
// Generic WMMA-f32 GEMM:  C[M,N] = act(A[M,K] @ W[N,K]^T + bias[N])
// Block tile 128x64, 8 wave32s; each wave owns a 16x64 strip (4 x v8f acc),
// so every A-fragment feeds a 4-deep WMMA burst. BK=32.
// Requires: M % 128 == 0, K % 32 == 0. FULL_N=false adds N guards (attn N=96).
// ---------------------------------------------------------------------------
template<bool FULL_N, int ACT>
__global__ void __launch_bounds__(256) gemm_bias_act(
    const float* __restrict__ A, int lda,
    const float* __restrict__ W,            // [N, K] row-major (torch Linear)
    const float* __restrict__ bias,
    float* __restrict__ C, int ldc,
    int M, int N, int K)
{
    __shared__ __align__(16) float As[128][36];
    __shared__ __align__(16) float Bs[64][36];

    const int tid   = threadIdx.x;
    const int lane  = tid & 31;
    const int wv    = tid >> 5;
    const int m0w   = wv * 16;              // wave M strip (8 waves x 16 = 128)
    const int mlane = lane & 15;
    const int klane = (lane >> 4) << 1;     // lanes 16-31 hold K+2,K+3
    const int mBase = blockIdx.y * 128;
    const int nBase = blockIdx.x * 64;

    v8f acc[4] = {};

    for (int kt = 0; kt < K; kt += 32) {
        // A tile: 128x32 = 1024 float4 -> 4 per thread
        #pragma unroll
        for (int q = tid; q < 1024; q += 256) {
            const int r = q >> 3, c = (q & 7) << 2;
            *(float4*)&As[r][c] =
                *(const float4*)(A + (size_t)(mBase + r) * lda + kt + c);
        }
        // B tile: 64x32 = 512 float4 -> 2 per thread
        #pragma unroll
        for (int q = tid; q < 512; q += 256) {
            const int r = q >> 3, c = (q & 7) << 2;
            if (FULL_N) {
                *(float4*)&Bs[r][c] =
                    *(const float4*)(W + (size_t)(nBase + r) * K + kt + c);
            } else {
                float4 bv = make_float4(0.f, 0.f, 0.f, 0.f);
                if (nBase + r < N)
                    bv = *(const float4*)(W + (size_t)(nBase + r) * K + kt + c);
                *(float4*)&Bs[r][c] = bv;
            }
        }
        if (kt + 32 < K)   // warm next A k-slab (global_prefetch_b8)
            __builtin_prefetch(A + (size_t)(mBase + (tid >> 1)) * lda + kt + 32, 0, 1);
        __syncthreads();

        #pragma unroll
        for (int kk = 0; kk < 8; ++kk) {
            const int kb = (kk << 2) + klane;
            v2f a; a.x = As[m0w + mlane][kb]; a.y = As[m0w + mlane][kb + 1];
            #pragma unroll
            for (int sub = 0; sub < 4; ++sub) {
                v2f b; b.x = Bs[sub * 16 + mlane][kb];
                       b.y = Bs[sub * 16 + mlane][kb + 1];
                acc[sub] = wmma4(a, b, acc[sub]);
            }
        }
        __syncthreads();
    }

    const int rowHi = (lane >> 4) * 8;      // lanes 16-31 hold rows M+8..M+15
    #pragma unroll
    for (int sub = 0; sub < 4; ++sub) {
        const int n = nBase + sub * 16 + mlane;
        if (FULL_N || n < N) {
            const float bv = bias[n];
            #pragma unroll
            for (int i = 0; i < 8; ++i) {
                const int m = mBase + m0w + rowHi + i;
                C[(size_t)m * ldc + n] = activate_<ACT>(acc[sub][i] + bv);
            }
        }
    }
}

// ---------------------------------------------------------------------------
// Fused encoder GRU (h0 = 0): for each row m of x[M_ENC,512] and col j of H:
//   acc_g[m,j] = sum_k x[m,k] * W_ih_e[g*H + j, k]      (3 gates share A-tile)
//   r = sig(accR + bir + bhr); z = sig(accZ + biz + bhz)
//   n = tanh(accN + bin + r*bhn);  h_enc = (1-z)*n
// Avoids materializing the 302 MB `gi` tensor.  Block 64x64, 6 acc/wave.
// ---------------------------------------------------------------------------
__global__ void __launch_bounds__(256) enc_gru_wmma(
    const float* __restrict__ X,            // [M_ENC, F_]
    const float* __restrict__ Wih,          // [3H, F_]
    const float* __restrict__ bih,          // [3H]
    const float* __restrict__ bhh,          // [3H]
    float* __restrict__ Hout)               // [M_ENC, H_]
{
    __shared__ __align__(16) float As[64][36];
    __shared__ __align__(16) float Bs[3][64][36];

    const int tid   = threadIdx.x;
    const int lane  = tid & 31;
    const int wv    = tid >> 5;
    const int m0w   = (wv & 3) * 16;
    const int n0w   = (wv >> 2) * 32;
    const int mlane = lane & 15;
    const int klane = (lane >> 4) << 1;
    const int mBase = blockIdx.y * 64;
    const int nBase = blockIdx.x * 64;      // column within H (0..255)

    v8f acc[3][2] = {};

    for (int kt = 0; kt < F_; kt += 32) {
        #pragma unroll
        for (int q = tid; q < 512; q += 256) {
            const int r = q >> 3, c = (q & 7) << 2;
            *(float4*)&As[r][c] =
                *(const float4*)(X + (size_t)(mBase + r) * F_ + kt + c);
            #pragma unroll
            for (int g = 0; g < 3; ++g)
                *(float4*)&Bs[g][r][c] =
                    *(const float4*)(Wih + (size_t)(g * H_ + nBase + r) * F_ + kt + c);
        }
        __syncthreads();

        #pragma unroll
        for (int kk = 0; kk < 8; ++kk) {
            const int kb = (kk << 2) + klane;
            v2f a; a.x = As[m0w + mlane][kb]; a.y = As[m0w + mlane][kb + 1];
            #pragma unroll
            for (int g = 0; g < 3; ++g) {
                v2f b0; b0.x = Bs[g][n0w + mlane][kb];
                        b0.y = Bs[g][n0w + mlane][kb + 1];
                acc[g][0] = wmma4(a, b0, acc[g][0]);
                v2f b1; b1.x = Bs[g][n0w + 16 + mlane][kb];
                        b1.y = Bs[g][n0w + 16 + mlane][kb + 1];
                acc[g][1] = wmma4(a, b1, acc[g][1]);
            }
        }
        __syncthreads();
    }

    const int rowHi = (lane >> 4) * 8;
    #pragma unroll
    for (int sub = 0; sub < 2; ++sub) {
        const int j   = nBase + n0w + sub * 16 + mlane;
        const float br  = bih[j]           + bhh[j];
        const float bz  = bih[j + H_]      + bhh[j + H_];
        const float bin = bih[j + 2 * H_];
        const float bhn = bhh[j + 2 * H_];
        #pragma unroll
        for (int i = 0; i < 8; ++i) {
            const int m = mBase + m0w + rowHi + i;
            const float r  = sigmoidf_(acc[0][sub][i] + br);
            const float zz = sigmoidf_(acc[1][sub][i] + bz);
            const float nn = tanhf(acc[2][sub][i] + bin + r * bhn);
            Hout[(size_t)m * H_ + j] = (1.0f - zz) * nn;
        }
    }
}

// ---------------------------------------------------------------------------
// applied[b,f] = sum_s aw[b,s] * enc_out[b,s,f]   (batched 1x96 @ 96x512)
// ---------------------------------------------------------------------------
__global__ void __launch_bounds__(128) attn_apply(
    const float* __restrict__ aw,           // [B_, S_IN]
    const float* __restrict__ enc_out,      // [B_, S_IN, F_]
    float* __restrict__ out)                // [B_, F_]
{
    __shared__ float s_aw[S_IN];
    const int b   = blockIdx.x;
    const int tid = threadIdx.x;
    if (tid < S_IN) s_aw[tid] = aw[(size_t)b * S_IN + tid];
    __syncthreads();
    const float* eb = enc_out + (size_t)b * S_IN * F_;
    for (int f = tid; f < F_; f += 128) {
        float acc = 0.0f;
        #pragma unroll 8
        for (int s = 0; s < S_IN; ++s)
            acc = fmaf(s_aw[s], eb[(size_t)s * F_ + f], acc);
        out[(size_t)b * F_ + f] = acc;
    }
}

// ---------------------------------------------------------------------------
// out[m, 0:512] = X[m,:]; out[m, 512:1024] = Y[m,:]
// ---------------------------------------------------------------------------
__global__ void __launch_bounds__(256) concat_rows(
    const float* __restrict__ X, const float* __restrict__ Y,
    float* __restrict__ out)
{
    const int idx = blockIdx.x * 256 + threadIdx.x;   // over B_*F_
    if (idx >= B_ * F_) return;
    const int m = idx >> 9, j = idx & (F_ - 1);
    out[(size_t)m * (2 * F_) + j]      = X[idx];
    out[(size_t)m * (2 * F_) + F_ + j] = Y[idx];
}

__global__ void __launch_bounds__(256) fill_zero(float* __restrict__ p, int n) {
    const int idx = blockIdx.x * 256 + threadIdx.x;
    if (idx < n) p[idx] = 0.0f;
}

// ---------------------------------------------------------------------------
// Decoder GRU gates + output + next input selection (teacher forcing flag
// is read from device memory at runtime).
// gi/gh already contain their biases (added in the GEMM epilogue).
// ---------------------------------------------------------------------------
__global__ void __launch_bounds__(256) gru_dec_ew(
    const float* __restrict__ gi,           // [B_, 3F_]
    const float* __restrict__ gh,           // [B_, 3F_]
    float* __restrict__ h,                  // [B_, F_] in/out
    const float* __restrict__ z_tar,        // [B_, S_PR, F_]
    const int* __restrict__ tf,
    float* __restrict__ dec_in,             // [B_, F_]
    float* __restrict__ out,                // [B_, S_PR, F_]
    int t)
{
    const int idx = blockIdx.x * 256 + threadIdx.x;
    if (idx >= B_ * F_) return;
    const int m = idx >> 9, j = idx & (F_ - 1);
    const size_t g0 = (size_t)m * (3 * F_);
    const float r  = sigmoidf_(gi[g0 + j]            + gh[g0 + j]);
    const float z  = sigmoidf_(gi[g0 + F_ + j]       + gh[g0 + F_ + j]);
    const float n  = tanhf(gi[g0 + 2 * F_ + j] + r *  gh[g0 + 2 * F_ + j]);
    const float h2 = (1.0f - z) * n + z * h[idx];
    h[idx] = h2;
    const size_t oidx = ((size_t)m * S_PR + t) * F_ + j;
    out[oidx]   = h2;
    dec_in[idx] = (*tf != 0) ? z_tar[oidx] : h2;
}

// ---------------------------------------------------------------------------
// Host-side GEMM dispatch (selects FULL_N / activation template instance)
// ---------------------------------------------------------------------------
static inline void run_gemm(const float* A, int lda, const float* W,
                            const float* bias, float* C, int ldc,
                            int M, int N, int K, int act, bool fullN,
                            hipStream_t s) {
    dim3 g((N + 63) / 64, M / 128);
    if (fullN) {
        if (act == 0)
            gemm_bias_act<true, 0><<<g, 256, 0, s>>>(A, lda, W, bias, C, ldc, M, N, K);
        else if (act == 1)
            gemm_bias_act<true, 1><<<g, 256, 0, s>>>(A, lda, W, bias, C, ldc, M, N, K);
        else
            gemm_bias_act<true, 2><<<g, 256, 0, s>>>(A, lda, W, bias, C, ldc, M, N, K);
    } else {
        if (act == 2)
            gemm_bias_act<false, 2><<<g, 256, 0, s>>>(A, lda, W, bias, C, ldc, M, N, K);
        else
            gemm_bias_act<false, 0><<<g, 256, 0, s>>>(A, lda, W, bias, C, ldc, M, N, K);
    }
}

// ---------------------------------------------------------------------------
// Host orchestration
// ---------------------------------------------------------------------------
extern "C" void kernel_launch(void* const* d_in, const int* in_sizes, int n_in,
                              void* d_out, int out_size, void* d_ws, size_t ws_size,
                              hipStream_t stream) {
    const float* z_in   = (const float*)d_in[0];
    const float* z_tar  = (const float*)d_in[1];
    const float* W_ih_e = (const float*)d_in[2];
    // d_in[3] = W_hh_e: unused (h0 == 0, only its bias matters)
    const float* b_ih_e = (const float*)d_in[4];
    const float* b_hh_e = (const float*)d_in[5];
    const float* W_fc   = (const float*)d_in[6];
    const float* b_fc   = (const float*)d_in[7];
    const float* W_attn = (const float*)d_in[8];
    const float* b_attn = (const float*)d_in[9];
    const float* W_comb = (const float*)d_in[10];
    const float* b_comb = (const float*)d_in[11];
    const float* W_ih_d = (const float*)d_in[12];
    const float* W_hh_d = (const float*)d_in[13];
    const float* b_ih_d = (const float*)d_in[14];
    const float* b_hh_d = (const float*)d_in[15];
    const int*   tf     = (const int*)d_in[16];
    float* out = (float*)d_out;

    // Workspace layout (floats)
    float* ws = (float*)d_ws;
    size_t off = 0;
    float* enc_out = ws + off; off += (size_t)M_ENC * F_;   // 192 MB
    float* h_enc   = ws + off; off += (size_t)M_ENC * H_;   //  96 MB
    float* h_dec   = ws + off; off += (size_t)B_ * F_;
    float* dec_in  = ws + off; off += (size_t)B_ * F_;
    float* cat     = ws + off; off += (size_t)B_ * 2 * F_;
    float* aw      = ws + off; off += (size_t)B_ * S_IN;
    float* applied = ws + off; off += (size_t)B_ * F_;
    float* comb    = ws + off; off += (size_t)B_ * F_;
    float* gi      = ws + off; off += (size_t)B_ * 3 * F_;
    float* gh      = ws + off; off += (size_t)B_ * 3 * F_;

    // --- Encoder ---
    enc_gru_wmma<<<dim3(H_ / 64, M_ENC / 64), 256, 0, stream>>>(
        z_in, W_ih_e, b_ih_e, b_hh_e, h_enc);

    // enc_out = relu(h_enc @ W_fc^T + b_fc)
    run_gemm(h_enc, H_, W_fc, b_fc, enc_out, F_, M_ENC, F_, H_,
             /*act=*/1, /*fullN=*/true, stream);

    // dec_h = h_enc[:, -1] @ W_fc^T + b_fc   (strided A = last timestep rows)
    run_gemm(h_enc + (size_t)(S_IN - 1) * H_, S_IN * H_,
             W_fc, b_fc, h_dec, F_, B_, F_, H_,
             /*act=*/0, /*fullN=*/true, stream);

    fill_zero<<<(B_ * F_ + 255) / 256, 256, 0, stream>>>(dec_in, B_ * F_);

    // --- Decoder scan ---
    for (int t = 0; t < S_PR; ++t) {
        concat_rows<<<(B_ * F_ + 255) / 256, 256, 0, stream>>>(dec_in, h_dec, cat);

        // aw = tanh([dec_in, h] @ W_attn^T + b_attn), N=96 (guarded tiles)
        run_gemm(cat, 2 * F_, W_attn, b_attn, aw, S_IN, B_, S_IN, 2 * F_,
                 /*act=*/2, /*fullN=*/false, stream);

        attn_apply<<<B_, 128, 0, stream>>>(aw, enc_out, applied);

        concat_rows<<<(B_ * F_ + 255) / 256, 256, 0, stream>>>(dec_in, applied, cat);

        // comb = relu([dec_in, applied] @ W_comb^T + b_comb)
        run_gemm(cat, 2 * F_, W_comb, b_comb, comb, F_, B_, F_, 2 * F_,
                 /*act=*/1, /*fullN=*/true, stream);

        // gi = comb @ W_ih_d^T + b_ih_d ; gh = h @ W_hh_d^T + b_hh_d
        run_gemm(comb, F_, W_ih_d, b_ih_d, gi, 3 * F_, B_, 3 * F_, F_,
                 /*act=*/0, /*fullN=*/true, stream);
        run_gemm(h_dec, F_, W_hh_d, b_hh_d, gh, 3 * F_, B_, 3 * F_, F_,
                 /*act=*/0, /*fullN=*/true, stream);

        gru_dec_ew<<<(B_ * F_ + 255) / 256, 256, 0, stream>>>(
            gi, gh, h_dec, z_tar, tf, dec_in, out, t);
    }
}